// DecodePredictions_82429012345279
// MI455X (gfx1250) — compile-verified
//
#include <hip/hip_runtime.h>

// ---------------- problem constants (from reference) ----------------
#define NANCH    76725
#define NCLS     90
#define ROWF     94          // 4 box + 90 class logits per anchor
#define PRE_K    1000
#define MPC      100         // MAX_PER_CLASS
#define MDET     100         // MAX_DET
#define NBATCH   8
#define CONF_THR 0.05f
#define IOU_THRS 0.5f
#define FINF __builtin_inff()

// gfx1250 async global->LDS path (guarded; falls back to plain loads)
#if __has_builtin(__builtin_amdgcn_global_load_async_to_lds_b64)
#define USE_ASYNC_LDS 1
#else
#define USE_ASYNC_LDS 0
#endif

#if USE_ASYNC_LDS
typedef int v2i32 __attribute__((vector_size(8)));
typedef __attribute__((address_space(1))) v2i32* gbl_v2i32_ptr;
typedef __attribute__((address_space(3))) v2i32* lds_v2i32_ptr;
#endif

// monotonic key: larger float  <=>  larger unsigned key
__device__ __forceinline__ unsigned key_of(float f) {
    unsigned u = __float_as_uint(f);
    return (u & 0x80000000u) ? ~u : (u | 0x80000000u);
}

// wave32 shuffle + LDS block argmax (ties -> smallest index; deterministic)
__device__ __forceinline__ void block_argmax(float v, int i,
                                             float* red_s, int* red_i,
                                             float& oV, int& oI)
{
    const int tid = (int)threadIdx.x;
#pragma unroll
    for (int off = 16; off > 0; off >>= 1) {
        float ov = __shfl_xor(v, off, 32);
        int   oi = __shfl_xor(i, off, 32);
        if (ov > v || (ov == v && oi < i)) { v = ov; i = oi; }
    }
    if ((tid & 31) == 0) { red_s[tid >> 5] = v; red_i[tid >> 5] = i; }
    __syncthreads();
    if (tid < 32) {
        const int nw = (int)(blockDim.x >> 5);
        float vv = (tid < nw) ? red_s[tid] : -FINF;
        int   ii = (tid < nw) ? red_i[tid] : 0x7fffffff;
#pragma unroll
        for (int off = 16; off > 0; off >>= 1) {
            float ov = __shfl_xor(vv, off, 32);
            int   oi = __shfl_xor(ii, off, 32);
            if (ov > vv || (ov == vv && oi < ii)) { vv = ov; ii = oi; }
        }
        if (tid == 0) { red_s[0] = vv; red_i[0] = ii; }
    }
    __syncthreads();
    oV = red_s[0]; oI = red_i[0];
    __syncthreads();
}

// ------------------------------------------------------------------
// Kernel 2 (defined first so the disasm snippet shows the gfx1250
// async global->LDS staging sequence): greedy NMS per (image,class).
// Candidate records staged into LDS with global_load_async_to_lds_b64
// + s_wait_asynccnt; boxes decoded on the fly from gathered reads.
// ------------------------------------------------------------------
__global__ __launch_bounds__(1024)
void nms_kernel(const float* __restrict__ pred,
                const float* __restrict__ anchors,
                const unsigned* __restrict__ cand,
                float* __restrict__ nmsOut)
{
    __shared__ unsigned rec[PRE_K * 2];   // 8 KB candidate records
    __shared__ float red_s[32];
    __shared__ int   red_i[32];
    __shared__ float selBox[5];           // x1,y1,x2,y2,area of winner

    const int blk = (int)blockIdx.x;      // b*NCLS + c
    const int b   = blk / NCLS;
    const int tid = (int)threadIdx.x;
    const unsigned* grec = cand + (size_t)blk * (PRE_K * 2);

#if USE_ASYNC_LDS
    if (tid < PRE_K) {
        __builtin_amdgcn_global_load_async_to_lds_b64(
            (gbl_v2i32_ptr)(grec + 2 * tid),
            (lds_v2i32_ptr)(rec + 2 * tid),
            0, 0);
    }
  #if __has_builtin(__builtin_amdgcn_s_wait_asynccnt)
    __builtin_amdgcn_s_wait_asynccnt(0);
  #else
    asm volatile("s_wait_asynccnt 0" ::: "memory");
  #endif
    __syncthreads();
#else
    if (tid < PRE_K) {
        rec[2 * tid]     = grec[2 * tid];
        rec[2 * tid + 1] = grec[2 * tid + 1];
    }
    __syncthreads();
#endif

    // decode my candidate (gathered reads for only 1000 anchors)
    float x1 = 0.f, y1 = 0.f, x2 = 0.f, y2 = 0.f, area = 0.f, sc = 0.f;
    bool act = false;
    if (tid < PRE_K) {
        const unsigned idx   = rec[2 * tid];
        const float    logit = __uint_as_float(rec[2 * tid + 1]);
        const float* p = pred + (size_t)b * NANCH * ROWF + (size_t)idx * ROWF;
        const float* a = anchors + (size_t)idx * 4;
        const float acx = a[0], acy = a[1], aw = a[2], ah = a[3];
        const float cx = p[0] * aw + acx;
        const float cy = p[1] * ah + acy;
        const float w  = expf(p[2]) * aw;
        const float h  = expf(p[3]) * ah;
        x1 = cx - 0.5f * w; y1 = cy - 0.5f * h;
        x2 = cx + 0.5f * w; y2 = cy + 0.5f * h;
        area = (x2 - x1) * (y2 - y1);
        sc   = 1.f / (1.f + expf(-logit));
        act  = sc > CONF_THR;
    }

    for (int m = 0; m < MPC; ++m) {
        float mv; int mi;
        block_argmax(act ? sc : -FINF, tid, red_s, red_i, mv, mi);
        const bool ok = mv > -FINF;
        float* o = nmsOut + ((size_t)blk * MPC + m) * 5;
        if (ok) {
            if (tid == mi) {
                o[0] = x1; o[1] = y1; o[2] = x2; o[3] = y2; o[4] = sc;
                selBox[0] = x1; selBox[1] = y1;
                selBox[2] = x2; selBox[3] = y2; selBox[4] = area;
            }
        } else if (tid == 0) {
            o[0] = 0.f; o[1] = 0.f; o[2] = 0.f; o[3] = 0.f; o[4] = -1.f;
        }
        __syncthreads();
        if (ok && act) {
            const float lx = fmaxf(x1, selBox[0]);
            const float ly = fmaxf(y1, selBox[1]);
            const float rx = fminf(x2, selBox[2]);
            const float ry = fminf(y2, selBox[3]);
            const float iw = fmaxf(rx - lx, 0.f);
            const float ih = fmaxf(ry - ly, 0.f);
            const float inter = iw * ih;
            const float iou = inter / (area + selBox[4] - inter + 1e-8f);
            if (iou > IOU_THRS) act = false;   // winner suppresses itself too (iou==1)
        }
        __syncthreads();
    }
}

// ------------------------------------------------------------------
// Kernel 1: per-(image,class) exact top-1000 by logit, via 3-level
// radix histogram refinement (10+10+12 bits -> exact 32-bit cutoff).
// One block per (b,c); L2 (192MB) keeps the 28.8MB per-image tensor
// resident across the strided passes.
// ------------------------------------------------------------------
__global__ __launch_bounds__(1024)
void topk_select_kernel(const float* __restrict__ pred, unsigned* __restrict__ cand)
{
    __shared__ int hist[4096];
    __shared__ int s_t0, s_t1, s_t2, s_above, s_need, s_nSel, s_nEq;

    const int blk = (int)blockIdx.x;            // b*NCLS + c
    const int b   = blk / NCLS;
    const int c   = blk % NCLS;
    const int tid = (int)threadIdx.x;
    const float* base = pred + (size_t)b * NANCH * ROWF + 4 + c;

    // ---- level 0: top 10 key bits
    for (int i = tid; i < 1024; i += 1024) hist[i] = 0;
    __syncthreads();
    for (int i = tid; i < NANCH; i += 1024) {
        if (i + 1024 < NANCH)
            __builtin_prefetch(base + (size_t)(i + 1024) * ROWF, 0, 1);
        unsigned k = key_of(base[(size_t)i * ROWF]);
        atomicAdd(&hist[k >> 22], 1);
    }
    __syncthreads();
    if (tid == 0) {
        int acc = 0, t = 0;
        for (int bin = 1023; bin >= 0; --bin) {
            if (acc + hist[bin] >= PRE_K) { t = bin; break; }
            acc += hist[bin];
        }
        s_t0 = t; s_above = acc;
    }
    __syncthreads();
    const unsigned t0 = (unsigned)s_t0;

    // ---- level 1: middle 10 bits among prefix matches
    for (int i = tid; i < 1024; i += 1024) hist[i] = 0;
    __syncthreads();
    for (int i = tid; i < NANCH; i += 1024) {
        unsigned k = key_of(base[(size_t)i * ROWF]);
        if ((k >> 22) == t0) atomicAdd(&hist[(k >> 12) & 1023u], 1);
    }
    __syncthreads();
    if (tid == 0) {
        int acc = s_above, t = 0;
        for (int bin = 1023; bin >= 0; --bin) {
            if (acc + hist[bin] >= PRE_K) { t = bin; break; }
            acc += hist[bin];
        }
        s_t1 = t; s_above = acc;
    }
    __syncthreads();
    const unsigned pre20 = (t0 << 10) | (unsigned)s_t1;

    // ---- level 2: low 12 bits among 20-bit prefix matches
    for (int i = tid; i < 4096; i += 1024) hist[i] = 0;
    __syncthreads();
    for (int i = tid; i < NANCH; i += 1024) {
        unsigned k = key_of(base[(size_t)i * ROWF]);
        if ((k >> 12) == pre20) atomicAdd(&hist[k & 4095u], 1);
    }
    __syncthreads();
    if (tid == 0) {
        int acc = s_above, t = 0;
        for (int bin = 4095; bin >= 0; --bin) {
            if (acc + hist[bin] >= PRE_K) { t = bin; break; }
            acc += hist[bin];
        }
        s_t2 = t; s_above = acc; s_need = PRE_K - acc;
        s_nSel = 0; s_nEq = 0;
    }
    __syncthreads();
    const unsigned V    = (pre20 << 12) | (unsigned)s_t2;  // exact 32-bit cutoff key
    const int      need = s_need;

    // ---- compaction: all keys > V, plus `need` keys == V
    unsigned* crec = cand + (size_t)blk * (PRE_K * 2);
    for (int i = tid; i < NANCH; i += 1024) {
        float    x = base[(size_t)i * ROWF];
        unsigned k = key_of(x);
        if (k > V) {
            int slot = atomicAdd(&s_nSel, 1);
            if (slot < PRE_K) {
                crec[2 * slot]     = (unsigned)i;
                crec[2 * slot + 1] = __float_as_uint(x);
            }
        } else if (k == V) {
            int r = atomicAdd(&s_nEq, 1);
            if (r < need) {
                int slot = atomicAdd(&s_nSel, 1);
                if (slot < PRE_K) {
                    crec[2 * slot]     = (unsigned)i;
                    crec[2 * slot + 1] = __float_as_uint(x);
                }
            }
        }
    }
}

// ------------------------------------------------------------------
// Kernel 3: per-image sorted top-100 over 90*100 per-class results.
// ------------------------------------------------------------------
__global__ __launch_bounds__(1024)
void finalize_kernel(const float* __restrict__ nmsOut, float* __restrict__ out)
{
    __shared__ float scl[NCLS * MPC];     // 36 KB scores
    __shared__ float red_s[32];
    __shared__ int   red_i[32];
    __shared__ int   sel[MDET];

    const int b   = (int)blockIdx.x;
    const int tid = (int)threadIdx.x;
    const float* src = nmsOut + (size_t)b * NCLS * MPC * 5;

    for (int i = tid; i < NCLS * MPC; i += 1024) scl[i] = src[(size_t)i * 5 + 4];
    __syncthreads();

    for (int j = 0; j < MDET; ++j) {
        float v = -FINF; int idx = NCLS * MPC;
        for (int i = tid; i < NCLS * MPC; i += 1024) {
            float s = scl[i];
            if (s > v) { v = s; idx = i; }   // ascending i -> first occurrence
        }
        float mv; int mi;
        block_argmax(v, idx, red_s, red_i, mv, mi);
        if (tid == 0) { sel[j] = mi; scl[mi] = -FINF; }
        __syncthreads();
    }

    float* outB = out;                         // [8][100][4]
    float* outS = out + NBATCH * MDET * 4;     // [8][100]
    float* outC = outS + NBATCH * MDET;        // [8][100]
    float* outV = outC + NBATCH * MDET;        // [8]

    if (tid < MDET) {
        const int e = sel[tid];
        const float* r = src + (size_t)e * 5;
        const float s = r[4];
        const bool keep = s > 0.f;
        const int o = b * MDET + tid;
        outS[o] = keep ? s : 0.f;
        outC[o] = keep ? (float)(e / MPC) : 0.f;
        outB[o * 4 + 0] = keep ? r[0] : 0.f;
        outB[o * 4 + 1] = keep ? r[1] : 0.f;
        outB[o * 4 + 2] = keep ? r[2] : 0.f;
        outB[o * 4 + 3] = keep ? r[3] : 0.f;
    }
    if (tid == 0) {
        int cnt = 0;
        for (int j = 0; j < MDET; ++j)
            if (src[(size_t)sel[j] * 5 + 4] > 0.f) cnt++;
        outV[b] = (float)cnt;
    }
}

// ------------------------------------------------------------------
extern "C" void kernel_launch(void* const* d_in, const int* in_sizes, int n_in,
                              void* d_out, int out_size, void* d_ws, size_t ws_size,
                              hipStream_t stream)
{
    (void)in_sizes; (void)n_in; (void)out_size; (void)ws_size;
    const float* pred    = (const float*)d_in[0];   // [8,76725,94]
    const float* anchors = (const float*)d_in[1];   // [76725,4]
    float* out = (float*)d_out;

    unsigned char* ws = (unsigned char*)d_ws;
    unsigned* cand = (unsigned*)ws;                                  // 720*1000*2 u32 = 5.76 MB
    float* nmsOut  = (float*)(ws + (size_t)NBATCH * NCLS * PRE_K * 2 * sizeof(unsigned)); // 720*100*5 f32

    topk_select_kernel<<<dim3(NBATCH * NCLS), dim3(1024), 0, stream>>>(pred, cand);
    nms_kernel       <<<dim3(NBATCH * NCLS), dim3(1024), 0, stream>>>(pred, anchors, cand, nmsOut);
    finalize_kernel  <<<dim3(NBATCH),        dim3(1024), 0, stream>>>(nmsOut, out);
}